// RGNN_71184787964473
// MI455X (gfx1250) — compile-verified
//
#include <hip/hip_runtime.h>
#include <math.h>

#define CIN  128
#define HID  256
#define OUTD 349
#define OUTP 352          // OUT padded to multiple of 16
#define NTYPES 4
#define ETYPES 5
#define N0v 120000
#define N1v 40000
#define N2v 10000
#define E0v 1500000
#define E1v 400000
#define K0TOT (ETYPES*CIN + NTYPES*CIN)   // 640 + 512 = 1152
#define K1TOT (ETYPES*HID + NTYPES*HID)   // 1280 + 1024 = 2304

typedef float v2f __attribute__((ext_vector_type(2)));
typedef float v8f __attribute__((ext_vector_type(8)));

// ---------------------------------------------------------------------------
// 1) Gather per-type input features: h0[n] = table[nt][min(li, rows-1)]
// ---------------------------------------------------------------------------
__global__ __launch_bounds__(256) void k_gather_input(
    const int* __restrict__ n_id, const int* __restrict__ node_type,
    const int* __restrict__ local_idx,
    const float* __restrict__ x0, const float* __restrict__ e1,
    const float* __restrict__ e2, const float* __restrict__ e3,
    float* __restrict__ h0, int* __restrict__ nt0)
{
    int idx = blockIdx.x * 256 + threadIdx.x;
    if (idx >= N0v * 32) return;
    int n  = idx >> 5;
    int c4 = idx & 31;
    int g  = n_id[n];
    int nt = node_type[g];
    int li = local_idx[g];
    const float* tab; int rows;
    switch (nt) {
        case 0:  tab = x0; rows = 100000; break;
        case 1:  tab = e1; rows = 50000;  break;
        case 2:  tab = e2; rows = 50000;  break;
        default: tab = e3; rows = 50000;  break;
    }
    int row = li < rows ? li : rows - 1;
    float4 v = ((const float4*)(tab + (size_t)row * CIN))[c4];
    ((float4*)(h0 + (size_t)n * CIN))[c4] = v;
    if (c4 == 0) nt0[n] = nt;
}

// ---------------------------------------------------------------------------
// 2) Edge scatter layer 0: one 32-lane subwave per edge, 4 floats/lane
// ---------------------------------------------------------------------------
__global__ __launch_bounds__(256) void k_scatter0(
    const int* __restrict__ src, const int* __restrict__ dst,
    const int* __restrict__ eid, const int* __restrict__ etype,
    const float* __restrict__ h0, float* __restrict__ F0, float* __restrict__ cnt)
{
    int e = blockIdx.x * 8 + (threadIdx.x >> 5);
    if (e >= E0v) return;
    int lane = threadIdx.x & 31;
    int s = src[e], d = dst[e], t = etype[eid[e]];
    float4 v = ((const float4*)(h0 + (size_t)s * CIN))[lane];
    float* p = F0 + (size_t)d * K0TOT + t * CIN + lane * 4;
    atomicAdd(p + 0, v.x); atomicAdd(p + 1, v.y);
    atomicAdd(p + 2, v.z); atomicAdd(p + 3, v.w);
    if (lane == 0) atomicAdd(cnt + d * ETYPES + t, 1.0f);
}

// ---------------------------------------------------------------------------
// 3) Finalize F0: divide sums by counts; append type-masked x (root term)
// ---------------------------------------------------------------------------
__global__ __launch_bounds__(256) void k_finalize0(
    float* __restrict__ F0, const float* __restrict__ cnt,
    const float* __restrict__ h0, const int* __restrict__ nt0)
{
    int idx = blockIdx.x * 256 + threadIdx.x;     // one float4 per thread
    if (idx >= N1v * (K0TOT / 4)) return;
    int n = idx / (K0TOT / 4);
    int q = idx - n * (K0TOT / 4);
    float4* rowp = (float4*)(F0 + (size_t)n * K0TOT);
    if (q < ETYPES * CIN / 4) {                   // mean region
        int t = q / (CIN / 4);
        float inv = 1.0f / fmaxf(cnt[n * ETYPES + t], 1.0f);
        float4 v = rowp[q];
        v.x *= inv; v.y *= inv; v.z *= inv; v.w *= inv;
        rowp[q] = v;
    } else {                                      // masked-x (root) region
        int q2 = q - ETYPES * CIN / 4;
        int t2 = q2 / (CIN / 4);
        int c4 = q2 & (CIN / 4 - 1);
        float4 v = make_float4(0.f, 0.f, 0.f, 0.f);
        if (nt0[n] == t2) v = ((const float4*)(h0 + (size_t)n * CIN))[c4];
        rowp[q] = v;
    }
}

// ---------------------------------------------------------------------------
// 4) WMMA f32 GEMM: C[M,Npad] = A[M,K] x [B1;B2][K,Nsrc->Npad] (+bias[nt], relu)
//    Block = 8 waves; each wave owns a 16-row strip x full Npad.
//    B panel (4 x Npad) staged K-transposed through double-buffered LDS so
//    each lane's (b.x,b.y) pair is one aligned ds_load_b64.
// ---------------------------------------------------------------------------
template<int NTILES>
__global__ __launch_bounds__(256) void k_gemm_wmma(
    const float* __restrict__ A, int M, int K,
    const float* __restrict__ B1, const float* __restrict__ B2,
    int ksplit, int Nsrc, int Npad,
    const float* __restrict__ bias, const int* __restrict__ ntab,
    float* __restrict__ C, int do_relu)
{
    extern __shared__ float ldsB[];               // 2 buffers of 4*Npad floats
    const int tid  = threadIdx.x;
    const int wave = tid >> 5;
    const int lane = tid & 31;
    const int half = lane >> 4;                   // lane half-wave
    const int lm   = lane & 15;
    const int mbase = blockIdx.x * 128 + wave * 16;
    const bool valid = (mbase < M);               // M is a multiple of 16

    v8f acc[NTILES];
#pragma unroll
    for (int t = 0; t < NTILES; ++t) acc[t] = {};

    // A 16x4 f32 layout: lanes 0-15 hold K{k,k+1}, lanes 16-31 hold K{k+2,k+3}
    const float* Arow = A + (size_t)(mbase + lm) * K + 2 * half;

    // stage rows kbase..kbase+3, K-transposed: buf[col*4 + (kr-kbase)]
    auto stage = [&](int kbase, float* buf) {
        for (int i = tid; i < 4 * Npad; i += 256) {
            int col = i >> 2;
            int kr  = kbase + (i & 3);
            float v = 0.0f;
            if (col < Nsrc)
                v = (kr < ksplit) ? B1[(size_t)kr * Nsrc + col]
                                  : B2[(size_t)(kr - ksplit) * Nsrc + col];
            buf[i] = v;
        }
    };

    stage(0, ldsB);
    __syncthreads();

    for (int k = 0; k < K; k += 4) {
        float* cur = ldsB + (((k >> 2) & 1)     ) * (4 * Npad);
        float* nxt = ldsB + ((((k >> 2) + 1) & 1)) * (4 * Npad);
        if (k + 4 < K) stage(k + 4, nxt);         // prefetch next panel

        if ((k & 127) == 0 && valid)              // feed the A stream ahead
            __builtin_prefetch(Arow + k + 128, 0, 0);

        v2f a;
        if (valid) { a.x = Arow[k]; a.y = Arow[k + 1]; }
        else       { a.x = 0.f;     a.y = 0.f; }

#pragma unroll
        for (int t = 0; t < NTILES; ++t) {
            // B 4x16 layout: lanes 0-15 K{k,k+1}, lanes 16-31 K{k+2,k+3};
            // transposed LDS makes this one contiguous 8B read per lane.
            v2f b = *(const v2f*)(cur + ((t * 16 + lm) << 2) + 2 * half);
            acc[t] = __builtin_amdgcn_wmma_f32_16x16x4_f32(
                false, a, false, b, (short)0, acc[t], false, false);
        }
        __syncthreads();
    }

    if (!valid) return;
#pragma unroll
    for (int t = 0; t < NTILES; ++t) {
#pragma unroll
        for (int r = 0; r < 8; ++r) {             // C VGPR r -> row r + 8*half
            int m   = mbase + r + 8 * half;
            int col = t * 16 + lm;
            float v = acc[t][r];
            if (col < Nsrc) v += bias[(size_t)ntab[m] * Nsrc + col];
            if (do_relu) v = fmaxf(v, 0.0f);
            C[(size_t)m * Npad + col] = v;
        }
    }
}

// ---------------------------------------------------------------------------
// 5) Edge scatter layer 1: 8 floats per lane (HID = 256)
// ---------------------------------------------------------------------------
__global__ __launch_bounds__(256) void k_scatter1(
    const int* __restrict__ src, const int* __restrict__ dst,
    const int* __restrict__ eid, const int* __restrict__ etype,
    const float* __restrict__ h1, float* __restrict__ F1, float* __restrict__ cnt)
{
    int e = blockIdx.x * 8 + (threadIdx.x >> 5);
    if (e >= E1v) return;
    int lane = threadIdx.x & 31;
    int s = src[e], d = dst[e], t = etype[eid[e]];
    const float4* hp = (const float4*)(h1 + (size_t)s * HID);
    float4 v0 = hp[lane * 2], v1 = hp[lane * 2 + 1];
    float* p = F1 + (size_t)d * K1TOT + t * HID + lane * 8;
    atomicAdd(p + 0, v0.x); atomicAdd(p + 1, v0.y);
    atomicAdd(p + 2, v0.z); atomicAdd(p + 3, v0.w);
    atomicAdd(p + 4, v1.x); atomicAdd(p + 5, v1.y);
    atomicAdd(p + 6, v1.z); atomicAdd(p + 7, v1.w);
    if (lane == 0) atomicAdd(cnt + d * ETYPES + t, 1.0f);
}

__global__ __launch_bounds__(256) void k_finalize1(
    float* __restrict__ F1, const float* __restrict__ cnt,
    const float* __restrict__ h1, const int* __restrict__ nt0)
{
    int idx = blockIdx.x * 256 + threadIdx.x;
    if (idx >= N2v * (K1TOT / 4)) return;
    int n = idx / (K1TOT / 4);
    int q = idx - n * (K1TOT / 4);
    float4* rowp = (float4*)(F1 + (size_t)n * K1TOT);
    if (q < ETYPES * HID / 4) {
        int t = q / (HID / 4);
        float inv = 1.0f / fmaxf(cnt[n * ETYPES + t], 1.0f);
        float4 v = rowp[q];
        v.x *= inv; v.y *= inv; v.z *= inv; v.w *= inv;
        rowp[q] = v;
    } else {
        int q2 = q - ETYPES * HID / 4;
        int t2 = q2 / (HID / 4);
        int c4 = q2 & (HID / 4 - 1);
        float4 v = make_float4(0.f, 0.f, 0.f, 0.f);
        if (nt0[n] == t2) v = ((const float4*)(h1 + (size_t)n * HID))[c4];
        rowp[q] = v;
    }
}

// ---------------------------------------------------------------------------
// 6) log_softmax: one wave per row, wave32 shuffle reductions
// ---------------------------------------------------------------------------
__global__ __launch_bounds__(256) void k_logsoftmax(
    const float* __restrict__ C1, float* __restrict__ out)
{
    int row = blockIdx.x * 8 + (threadIdx.x >> 5);
    if (row >= N2v) return;
    int lane = threadIdx.x & 31;
    const float* x = C1 + (size_t)row * OUTP;
    float mx = -3.402823466e+38f;
    for (int c = lane; c < OUTD; c += 32) mx = fmaxf(mx, x[c]);
    for (int o = 16; o; o >>= 1) mx = fmaxf(mx, __shfl_xor(mx, o, 32));
    float s = 0.0f;
    for (int c = lane; c < OUTD; c += 32) s += expf(x[c] - mx);
    for (int o = 16; o; o >>= 1) s += __shfl_xor(s, o, 32);
    float lg = logf(s) + mx;
    float* y = out + (size_t)row * OUTD;
    for (int c = lane; c < OUTD; c += 32) y[c] = x[c] - lg;
}

// ---------------------------------------------------------------------------
extern "C" void kernel_launch(void* const* d_in, const int* in_sizes, int n_in,
                              void* d_out, int out_size, void* d_ws, size_t ws_size,
                              hipStream_t stream) {
    const int*   n_id      = (const int*)  d_in[0];
    const float* x0        = (const float*)d_in[1];
    const int*   src0      = (const int*)  d_in[2];
    const int*   dst0      = (const int*)  d_in[3];
    const int*   e_id0     = (const int*)  d_in[4];
    const int*   src1      = (const int*)  d_in[5];
    const int*   dst1      = (const int*)  d_in[6];
    const int*   e_id1     = (const int*)  d_in[7];
    const int*   edge_type = (const int*)  d_in[8];
    const int*   node_type = (const int*)  d_in[9];
    const int*   local_idx = (const int*)  d_in[10];
    const float* emb1      = (const float*)d_in[11];
    const float* emb2      = (const float*)d_in[12];
    const float* emb3      = (const float*)d_in[13];
    const float* relW0     = (const float*)d_in[14];
    const float* rootW0    = (const float*)d_in[15];
    const float* rootb0    = (const float*)d_in[16];
    const float* relW1     = (const float*)d_in[17];
    const float* rootW1    = (const float*)d_in[18];
    const float* rootb1    = (const float*)d_in[19];
    float* out = (float*)d_out;

    // workspace carve-up (256B aligned)
    char* ws = (char*)d_ws;
    size_t off = 0;
    auto carve = [&](size_t bytes) -> char* {
        char* p = ws + off; off += (bytes + 255) & ~(size_t)255; return p;
    };
    float* h0   = (float*)carve((size_t)N0v * CIN   * 4);
    int*   nt0  = (int*)  carve((size_t)N0v         * 4);
    float* F0   = (float*)carve((size_t)N1v * K0TOT * 4);
    float* cnt0 = (float*)carve((size_t)N1v * ETYPES* 4);
    float* h1   = (float*)carve((size_t)N1v * HID   * 4);
    float* F1   = (float*)carve((size_t)N2v * K1TOT * 4);
    float* cnt1 = (float*)carve((size_t)N2v * ETYPES* 4);
    float* C1   = (float*)carve((size_t)N2v * OUTP  * 4);
    (void)ws_size; (void)in_sizes; (void)n_in; (void)out_size;

    // zero accumulation buffers (graph-capture-safe async memsets)
    hipMemsetAsync(F0,   0, (size_t)N1v * K0TOT  * 4, stream);
    hipMemsetAsync(cnt0, 0, (size_t)N1v * ETYPES * 4, stream);
    hipMemsetAsync(F1,   0, (size_t)N2v * K1TOT  * 4, stream);
    hipMemsetAsync(cnt1, 0, (size_t)N2v * ETYPES * 4, stream);

    // 1) gather input features
    k_gather_input<<<(N0v * 32 + 255) / 256, 256, 0, stream>>>(
        n_id, node_type, local_idx, x0, emb1, emb2, emb3, h0, nt0);

    // 2-3) layer-0 aggregation + finalize
    k_scatter0<<<(E0v + 7) / 8, 256, 0, stream>>>(
        src0, dst0, e_id0, edge_type, h0, F0, cnt0);
    k_finalize0<<<(N1v * (K0TOT / 4) + 255) / 256, 256, 0, stream>>>(
        F0, cnt0, h0, nt0);

    // 4) layer-0 GEMM (WMMA f32): h1 = relu(F0 @ [relW0;rootW0] + rootb0[nt])
    k_gemm_wmma<HID / 16><<<(N1v + 127) / 128, 256, 2 * 4 * HID * 4, stream>>>(
        F0, N1v, K0TOT, relW0, rootW0, ETYPES * CIN, HID, HID,
        rootb0, nt0, h1, 1);

    // 5) layer-1 aggregation + finalize
    k_scatter1<<<(E1v + 7) / 8, 256, 0, stream>>>(
        src1, dst1, e_id1, edge_type, h1, F1, cnt1);
    k_finalize1<<<(N2v * (K1TOT / 4) + 255) / 256, 256, 0, stream>>>(
        F1, cnt1, h1, nt0);

    // 6) layer-1 GEMM (WMMA f32): C1 = F1 @ [relW1;rootW1] + rootb1[nt]
    k_gemm_wmma<OUTP / 16><<<(N2v + 127) / 128, 256, 2 * 4 * OUTP * 4, stream>>>(
        F1, N2v, K1TOT, relW1, rootW1, ETYPES * HID, OUTD, OUTP,
        rootb1, nt0, C1, 0);

    // 7) log_softmax -> output
    k_logsoftmax<<<(N2v + 7) / 8, 256, 0, stream>>>(C1, out);
}